// AvgPoolNd_39573828665511
// MI455X (gfx1250) — compile-verified
//
#include <hip/hip_runtime.h>

// ---------------------------------------------------------------------------
// 4-D average pool, kernel=stride=2, over [2,16,32,32,32,32] f32.
// Memory-bound (128 MiB in / 8 MiB out, ~6 us floor at 23.3 TB/s).
// CDNA5 path: async global->LDS double-buffered pipeline + NT stores.
// ---------------------------------------------------------------------------

#if defined(__has_builtin)
#if __has_builtin(__builtin_amdgcn_global_load_async_to_lds_b128)
#define HAVE_ASYNC_LDS 1
#else
#define HAVE_ASYNC_LDS 0
#endif
#if __has_builtin(__builtin_amdgcn_s_wait_asynccnt)
#define WAIT_ASYNC(n) __builtin_amdgcn_s_wait_asynccnt(n)
#else
#define WAIT_ASYNC(n) asm volatile("s_wait_asynccnt %0" ::"i"(n) : "memory")
#endif
#else
#define HAVE_ASYNC_LDS 0
#define WAIT_ASYNC(n)
#endif

// Builtin signature (from hipcc diagnostic): pointers to v4i32 in
// global (AS1) and LDS (AS3) address spaces.
typedef int v4i __attribute__((vector_size(4 * sizeof(int))));
typedef __attribute__((address_space(1))) v4i GV4;
typedef __attribute__((address_space(3))) v4i LV4;

// Stage one 16 KiB tile (two 8 KiB contiguous chunks, d1=2*o1 and 2*o1+1)
// into LDS. 256 threads x 4 x 16B = 16 KiB. Per wave: 4 async ops in flight.
__device__ __forceinline__ void stage_tile(const float* __restrict__ gbase,
                                           float* lbuf, int tid) {
  const int half = tid >> 7;   // which d1 row (chunk)
  const int idx  = tid & 127;  // position within 8 KiB chunk
  const float* g = gbase + half * 32768 + idx * 4;  // 32768 = d1 stride (floats)
  float*       l = lbuf + half * 2048 + idx * 4;
#if HAVE_ASYNC_LDS
#pragma unroll
  for (int j = 0; j < 4; ++j) {
    __builtin_amdgcn_global_load_async_to_lds_b128(
        (GV4*)(g + j * 512), (LV4*)(l + j * 512), 0, 0);
  }
#else
#pragma unroll
  for (int j = 0; j < 4; ++j)
    *(float4*)(l + j * 512) = *(const float4*)(g + j * 512);
#endif
}

__global__ __launch_bounds__(256) void avgpool4d_k2s2(
    const float* __restrict__ in, float* __restrict__ out) {
  __shared__ float smem[8192];  // 2 x 16 KiB tile buffers (32 KiB total)

  const int tid = threadIdx.x;      // 0..255
  const int wg  = blockIdx.x;       // 0..511 : (b*16+c)*16 + o1
  const int bc  = wg >> 4;          // 0..31
  const int o1  = wg & 15;          // 0..15

  // input:  idx = ((bc*32 + d1)*32 + d2)*1024 + d3*32 + d4
  const float* inBase = in + (size_t)bc * (32u * 32u * 1024u)
                           + (size_t)(2 * o1) * (32u * 1024u);
  // output: idx = ((bc*16 + o1)*16 + o2)*256 + o3*16 + o4
  float* outBase = out + (size_t)(bc * 16 + o1) * 4096u;

  const int o3 = tid >> 4;
  const int o4 = tid & 15;

  // Prefetch tile 0 (o2 = 0): d2 pair stride = 2 planes = 2048 floats.
  stage_tile(inBase, smem, tid);

  for (int t = 0; t < 16; ++t) {
    const int cur = t & 1;
    if (t + 1 < 16) {
      // Prefetch next tile into the other buffer, then wait until only the
      // 4 just-issued ops remain (async loads complete in order per wave),
      // i.e. tile t's data is fully in LDS.
      stage_tile(inBase + (size_t)(t + 1) * 2048u, smem + ((t + 1) & 1) * 4096, tid);
#if HAVE_ASYNC_LDS
      WAIT_ASYNC(4);
#endif
    } else {
#if HAVE_ASYNC_LDS
      WAIT_ASYNC(0);
#endif
    }
    __syncthreads();  // all waves' LDS data visible

    // 2x2x2x2 window: i = d1 half, j = d2 plane, k = d3 row, float2 = d4 pair.
    const float* buf = smem + cur * 4096 + (2 * o3) * 32 + 2 * o4;
    float s = 0.0f;
#pragma unroll
    for (int i = 0; i < 2; ++i)
#pragma unroll
      for (int j = 0; j < 2; ++j)
#pragma unroll
        for (int k = 0; k < 2; ++k) {
          const float2 v = *(const float2*)(buf + i * 2048 + j * 1024 + k * 32);
          s += v.x + v.y;
        }

    // Write-once output stream: non-temporal, coalesced 1 KiB per WG per tile.
    __builtin_nontemporal_store(s * 0.0625f, outBase + t * 256 + tid);

    __syncthreads();  // protect buf[cur] before tile t+2 prefetch overwrites it
  }
}

extern "C" void kernel_launch(void* const* d_in, const int* in_sizes, int n_in,
                              void* d_out, int out_size, void* d_ws, size_t ws_size,
                              hipStream_t stream) {
  (void)in_sizes; (void)n_in; (void)out_size; (void)d_ws; (void)ws_size;
  const float* in = (const float*)d_in[0];
  float* out = (float*)d_out;
  // 512 workgroups = (B*C=32) x (o1=16); 256 threads = 8 wave32s each.
  avgpool4d_k2s2<<<dim3(512), dim3(256), 0, stream>>>(in, out);
}